// TrendLoss_15513421873612
// MI455X (gfx1250) — compile-verified
//
#include <hip/hip_runtime.h>
#include <math.h>

typedef __attribute__((ext_vector_type(2))) float v2f;
typedef __attribute__((ext_vector_type(8))) float v8f;

// Optional CDNA5 async global->LDS path (guarded so either toolchain compiles)
#if defined(__has_builtin)
#if __has_builtin(__builtin_amdgcn_global_load_async_to_lds_b32)
#define HAVE_ASYNC_LDS 1
#endif
#if __has_builtin(__builtin_amdgcn_s_wait_asynccnt)
#define HAVE_WAIT_ASYNC 1
#endif
#endif

typedef __attribute__((address_space(1))) int gint_t;   // global int
typedef __attribute__((address_space(3))) int lint_t;   // LDS int

// Problem constants (from reference setup_inputs)
#define NT      3650
#define NGAGE   2000
#define NY      5
#define NCOL    (NGAGE * NY)      // 10000
#define NYEARS  10
#define NDAYS   365
#define NPAIR   45                // 10 choose 2
#define MEDIDX  22                // (45-1)//2

// Launch geometry
#define NBLK_A  1024              // rmse partial blocks
#define THR_A   256
#define COLS_PER_WG 16
#define THR_B   256
#define NTILES  (NCOL / COLS_PER_WG)          // 625
#define NBLK_B  (2 * NYEARS * NTILES)         // 12500
#define THR_C   256
#define NBLK_C  ((NCOL + THR_C - 1) / THR_C)  // 40

// ---------------------------------------------------------------------------
// Kernel A: per-variable masked squared-error + valid-count block partials.
// Each thread handles 20 consecutive floats (lcm of float4 and the 5-variable
// period) so every load is b128 and every variable index is compile-time.
// ---------------------------------------------------------------------------
__global__ void rmse_partial_kernel(const float* __restrict__ outp,
                                    const float* __restrict__ targ,
                                    float* __restrict__ partials) {
    const int NGROUP20 = (NT * NCOL) / 20;   // 1,825,000 groups of 20
    float se[NY]  = {0.f, 0.f, 0.f, 0.f, 0.f};
    float cnt[NY] = {0.f, 0.f, 0.f, 0.f, 0.f};
    int stride = gridDim.x * blockDim.x;

#define ACC1(P, TE, OE)                                        \
    {                                                          \
        float t = (TE);                                        \
        float o = (OE);                                        \
        bool valid = (t == t);                                 \
        float d = o - t;                                       \
        se[(P) % NY]  += valid ? d * d : 0.0f;                 \
        cnt[(P) % NY] += valid ? 1.0f  : 0.0f;                 \
    }

    for (int g = blockIdx.x * blockDim.x + threadIdx.x; g < NGROUP20; g += stride) {
        size_t base = (size_t)g * 20;                 // 80B-aligned
        const float4* t4 = (const float4*)(targ + base);
        const float4* o4 = (const float4*)(outp + base);
#pragma unroll
        for (int q = 0; q < 5; ++q) {
            float4 tv = t4[q];
            float4 ov = o4[q];
            ACC1(q * 4 + 0, tv.x, ov.x);
            ACC1(q * 4 + 1, tv.y, ov.y);
            ACC1(q * 4 + 2, tv.z, ov.z);
            ACC1(q * 4 + 3, tv.w, ov.w);
        }
    }
#undef ACC1

    __shared__ float red[THR_A];
    int tid = threadIdx.x;
#pragma unroll
    for (int slot = 0; slot < 2 * NY; ++slot) {
        int v = slot % NY;
        red[tid] = (slot < NY) ? se[v] : cnt[v];
        __syncthreads();
        for (int s = THR_A / 2; s > 0; s >>= 1) {
            if (tid < s) red[tid] += red[tid + s];
            __syncthreads();
        }
        if (tid == 0) partials[blockIdx.x * (2 * NY) + slot] = red[0];
        __syncthreads();
    }
}

// ---------------------------------------------------------------------------
// Kernel B: per (tensor, year, 16-column tile):
//   - stage 365 days x 16 cols into LDS (async global->LDS when available)
//   - mean over 365 days via V_WMMA_F32_16X16X4_F32 with all-ones B (row sums)
//   - workgroup-cooperative bitonic sort (512) of all 16 columns in parallel
//   - emit mean + 5 order statistics into statBuf[tensor][stat][year][col]
// ---------------------------------------------------------------------------
__global__ void year_stats_kernel(const float* __restrict__ outp,
                                  const float* __restrict__ targ,
                                  float* __restrict__ statBuf) {
    __shared__ float colsS[COLS_PER_WG][520];   // 520 stride: bank-conflict pad

    int b      = blockIdx.x;
    int tensor = b / (NYEARS * NTILES);         // 0 = output(pred), 1 = target
    int rem    = b % (NYEARS * NTILES);
    int year   = rem / NTILES;
    int tile   = rem % NTILES;
    int colBase = tile * COLS_PER_WG;

    const float* __restrict__ src = (tensor == 0) ? outp : targ;

    int tid = threadIdx.x;
    int c  = tid & 15;       // column within tile
    int dr = tid >> 4;       // 0..15

    size_t baseIdx = (size_t)(year * NDAYS) * NCOL + colBase + c;

#if HAVE_ASYNC_LDS
    // CDNA5 async copy: global -> LDS without VGPR round trip (ASYNCcnt)
    for (int day = dr; day < NDAYS; day += 16) {
        __builtin_amdgcn_global_load_async_to_lds_b32(
            (gint_t*)(src + baseIdx + (size_t)day * NCOL),
            (lint_t*)&colsS[c][day],
            0, 0);
    }
#if HAVE_WAIT_ASYNC
    __builtin_amdgcn_s_wait_asynccnt(0);
#else
    asm volatile("s_wait_asynccnt 0" ::: "memory");
#endif
#else
    for (int day = dr; day < NDAYS; day += 16)
        colsS[c][day] = src[baseIdx + (size_t)day * NCOL];
#endif
    // pad region for the bitonic sort
    for (int day = NDAYS + dr; day < 512; day += 16)
        colsS[c][day] = __builtin_inff();
    __syncthreads();

    // --- means via fp32 WMMA: D = A(16x4) * ones(4x16), accumulate over K ---
    if (tid < 32) {
        int lane = tid;
        int col  = lane & 15;
        int koff = (lane < 16) ? 0 : 2;   // A layout: VGPR0 K=0|2, VGPR1 K=1|3
        v8f acc = {0.f, 0.f, 0.f, 0.f, 0.f, 0.f, 0.f, 0.f};
        v2f ones;
        ones.x = 1.0f; ones.y = 1.0f;
        for (int k = 0; k < 364; k += 4) {
            v2f a;
            a.x = colsS[col][k + koff];
            a.y = colsS[col][k + koff + 1];
            acc = __builtin_amdgcn_wmma_f32_16x16x4_f32(
                false, a, false, ones, (short)0, acc, false, false);
        }
        // D column N=0: lane 0 holds M=0..7 in acc[0..7]; lane 16 holds M=8..15
        if (lane == 0) {
#pragma unroll
            for (int r = 0; r < 8; ++r) {
                float sum = acc[r] + colsS[r][364];   // remainder day
                statBuf[(((size_t)tensor * 6 + 0) * NYEARS + year) * NCOL
                        + colBase + r] = sum * (1.0f / 365.0f);
            }
        } else if (lane == 16) {
#pragma unroll
            for (int r = 0; r < 8; ++r) {
                float sum = acc[r] + colsS[r + 8][364];
                statBuf[(((size_t)tensor * 6 + 0) * NYEARS + year) * NCOL
                        + colBase + r + 8] = sum * (1.0f / 365.0f);
            }
        }
    }
    __syncthreads();

    // --- bitonic sort, 16 columns in parallel, ascending ---
    for (int k = 2; k <= 512; k <<= 1) {
        for (int j = k >> 1; j > 0; j >>= 1) {
            for (int p = tid; p < COLS_PER_WG * 256; p += THR_B) {
                int cc = p >> 8;
                int q  = p & 255;
                int i  = ((q & ~(j - 1)) << 1) | (q & (j - 1));
                int l  = i | j;
                bool up = ((i & k) == 0);
                float A = colsS[cc][i];
                float B = colsS[cc][l];
                if (up ? (A > B) : (A < B)) {
                    colsS[cc][i] = B;
                    colsS[cc][l] = A;
                }
            }
            __syncthreads();
        }
    }

    // --- order statistics (ranks from PercentLst over 365 valid entries) ---
    if (tid < 5 * COLS_PER_WG) {
        const int QIDX[5] = {364, 357, 182, 109, 7};
        int s  = tid >> 4;        // 0..4
        int cc = tid & 15;
        statBuf[(((size_t)tensor * 6 + 1 + s) * NYEARS + year) * NCOL
                + colBase + cc] = colsS[cc][QIDX[s]];
    }
}

// ---------------------------------------------------------------------------
// Kernel C: Theil-Sen lower-median slope across 10 years per (col, stat),
// trend accumulation and block-partial reduction.
// ---------------------------------------------------------------------------
__device__ float theil_sen_10(const float* __restrict__ statBuf,
                              int tensor, int s, int col) {
    float y[NYEARS];
#pragma unroll
    for (int yr = 0; yr < NYEARS; ++yr)
        y[yr] = statBuf[(((size_t)tensor * 6 + s) * NYEARS + yr) * NCOL + col];
    float d[NPAIR];
    int m = 0;
    for (int i = 0; i < NYEARS - 1; ++i)
        for (int j = i + 1; j < NYEARS; ++j)
            d[m++] = (y[j] - y[i]) / (float)(j - i);
    // insertion sort, then lower median (sorted index 22 of 45)
    for (int a = 1; a < NPAIR; ++a) {
        float key = d[a];
        int bb = a - 1;
        while (bb >= 0 && d[bb] > key) { d[bb + 1] = d[bb]; --bb; }
        d[bb + 1] = key;
    }
    return d[MEDIDX];
}

__global__ void theilsen_kernel(const float* __restrict__ statBuf,
                                float* __restrict__ trendPartial) {
    int col = blockIdx.x * blockDim.x + threadIdx.x;
    float trend = 0.0f;
    if (col < NCOL) {
        for (int s = 0; s < 6; ++s) {   // stat 0 = mean, 1..5 = quantiles
            float tsP = theil_sen_10(statBuf, 0, s, col);
            float tsT = theil_sen_10(statBuf, 1, s, col);
            float dlt = tsT - tsP;
            trend += dlt * dlt;
        }
    }
    __shared__ float red[THR_C];
    int tid = threadIdx.x;
    red[tid] = trend;
    __syncthreads();
    for (int s = THR_C / 2; s > 0; s >>= 1) {
        if (tid < s) red[tid] += red[tid + s];
        __syncthreads();
    }
    if (tid == 0) trendPartial[blockIdx.x] = red[0];
}

// ---------------------------------------------------------------------------
// Kernel D: combine RMSE partials + trend partials into the scalar loss.
// ---------------------------------------------------------------------------
__global__ void finalize_kernel(const float* __restrict__ rmsePartials,
                                const float* __restrict__ trendPartial,
                                float* __restrict__ out) {
    if (blockIdx.x == 0 && threadIdx.x == 0) {
        float loss = 0.0f;
        for (int v = 0; v < NY; ++v) {
            float se = 0.0f, cnt = 0.0f;
            for (int b = 0; b < NBLK_A; ++b) {
                se  += rmsePartials[b * (2 * NY) + v];
                cnt += rmsePartials[b * (2 * NY) + NY + v];
            }
            loss += sqrtf(se / cnt);
        }
        float tr = 0.0f;
        for (int b = 0; b < NBLK_C; ++b) tr += trendPartial[b];
        loss += tr / (float)NGAGE;
        out[0] = loss;
    }
}

// ---------------------------------------------------------------------------
extern "C" void kernel_launch(void* const* d_in, const int* in_sizes, int n_in,
                              void* d_out, int out_size, void* d_ws, size_t ws_size,
                              hipStream_t stream) {
    const float* outp = (const float*)d_in[0];   // "output" (prediction)
    const float* targ = (const float*)d_in[1];   // "target"
    float* out = (float*)d_out;

    // workspace layout (floats)
    float* ws = (float*)d_ws;
    float* rmsePartials = ws;                                   // 1024*10
    float* statBuf      = rmsePartials + NBLK_A * 2 * NY;       // 2*6*10*10000
    float* trendPartial = statBuf + (size_t)2 * 6 * NYEARS * NCOL; // 40

    rmse_partial_kernel<<<NBLK_A, THR_A, 0, stream>>>(outp, targ, rmsePartials);
    year_stats_kernel<<<NBLK_B, THR_B, 0, stream>>>(outp, targ, statBuf);
    theilsen_kernel<<<NBLK_C, THR_C, 0, stream>>>(statBuf, trendPartial);
    finalize_kernel<<<1, 64, 0, stream>>>(rmsePartials, trendPartial, out);
}